// UPSSITS_39350490366325
// MI455X (gfx1250) — compile-verified
//
#include <hip/hip_runtime.h>
#include <hip/hip_bf16.h>

// Problem constants (from reference)
#define BB   256
#define CC   64
#define TT   406
#define DD   10
#define TD   4060          // T*D
#define KD   4468          // TD + T padded to multiple of 4 (4466 -> 4468)
#define NKT  (KD / 4)      // 1117 k-tiles of 4

// Output layout (floats), concatenated in reference return order
#define OFF_OUTSEQ  0u
#define OFF_INSEQ   1039360u
#define OFF_LOSS    2078720u
#define OFF_IDX     2095104u
#define OFF_LAB     2095360u
#define OFF_MASK    2095616u
#define OFF_PCOUNT  2199552u

// Workspace layout (floats)
#define WS_WN     0u          // 4060
#define WS_TERMA  4096u       // 256
#define WS_A      4352u       // 256*4468 = 1,143,808
#define WS_B      1148160u    // 4468*64  =   285,952  (fragment-staged)

typedef float v2f __attribute__((ext_vector_type(2)));
typedef float v8f __attribute__((ext_vector_type(8)));

// ---------------------------------------------------------------------------
// K0: wn = softplus(w) / sum(softplus(w)), single block
// ---------------------------------------------------------------------------
__global__ void k0_softplus_norm(const float* __restrict__ w, float* __restrict__ wn) {
    __shared__ float red[256];
    __shared__ float inv;
    float s = 0.f;
    for (int i = threadIdx.x; i < TD; i += 256) {
        float v  = w[i];
        float sp = fmaxf(v, 0.f) + log1pf(expf(-fabsf(v)));   // stable softplus
        wn[i] = sp;
        s += sp;
    }
    red[threadIdx.x] = s;
    __syncthreads();
    for (int off = 128; off > 0; off >>= 1) {
        if (threadIdx.x < off) red[threadIdx.x] += red[threadIdx.x + off];
        __syncthreads();
    }
    if (threadIdx.x == 0) inv = 1.0f / red[0];
    __syncthreads();
    for (int i = threadIdx.x; i < TD; i += 256) wn[i] *= inv;
}

// ---------------------------------------------------------------------------
// K1: A[b, col] = x*m*w for col<TD ; m for col in [TD,TD+T) ; 0 pad. Row-major.
// ---------------------------------------------------------------------------
__global__ void k1_build_a(const float* __restrict__ x, const float* __restrict__ mask,
                           const float* __restrict__ wn, float* __restrict__ A) {
    int idx = blockIdx.x * blockDim.x + threadIdx.x;
    if (idx >= BB * KD) return;
    int b = idx / KD, col = idx - b * KD;
    float v = 0.f;
    if (col < TD) {
        int t = col / DD;
        v = x[b * TD + col] * mask[b * TT + t] * wn[col];
    } else if (col < TD + TT) {
        v = mask[b * TT + (col - TD)];
    }
    A[idx] = v;
}

// ---------------------------------------------------------------------------
// K2: Bmat[k,p] = -2*proto[p,k] (k<TD) ; q[t,p] (k in [TD,TD+T)) ; 0 pad.
// Written directly in per-(ptile, ktile) fragment-staged order so the GEMM
// wave issues one coalesced float2 per lane:
//   Bst[((pt*NKT + kt)*32 + hi*16 + n)*2 + j] = Bmat[kt*4 + 2*hi + j][pt*16 + n]
// ---------------------------------------------------------------------------
__global__ void k2_build_b(const float* __restrict__ proto, const float* __restrict__ wn,
                           float* __restrict__ Bst) {
    int idx = blockIdx.x * blockDim.x + threadIdx.x;
    if (idx >= KD * CC) return;
    int k = idx / CC, p = idx - k * CC;
    float v = 0.f;
    if (k < TD) {
        v = -2.0f * proto[p * TD + k];
    } else if (k < TD + TT) {
        int t = k - TD;
        float s = 0.f;
        #pragma unroll
        for (int d = 0; d < DD; ++d) {
            float pv = proto[p * TD + t * DD + d];
            s += wn[t * DD + d] * pv * pv;
        }
        v = s;
    }
    int kt = k >> 2, kr = k & 3, hi = kr >> 1, j = kr & 1;
    int pt = p >> 4, n = p & 15;
    Bst[(((unsigned)(pt * NKT + kt) * 32u) + (unsigned)(hi * 16 + n)) * 2u + (unsigned)j] = v;
}

// ---------------------------------------------------------------------------
// K3: term_a[b] = sum_td A[b,td] * x[b,td]   (A already = x*m*w there)
// ---------------------------------------------------------------------------
__global__ void k3_term_a(const float* __restrict__ x, const float* __restrict__ A,
                          float* __restrict__ terma) {
    __shared__ float red[256];
    int b = blockIdx.x;
    float s = 0.f;
    for (int i = threadIdx.x; i < TD; i += 256)
        s += A[(size_t)b * KD + i] * x[(size_t)b * TD + i];
    red[threadIdx.x] = s;
    __syncthreads();
    for (int off = 128; off > 0; off >>= 1) {
        if (threadIdx.x < off) red[threadIdx.x] += red[threadIdx.x + off];
        __syncthreads();
    }
    if (threadIdx.x == 0) terma[b] = red[0];
}

// ---------------------------------------------------------------------------
// K4: f32 WMMA GEMM: loss[256x64] = A[256xKD] * Bmat[KDx64] + term_a[b]
// One wave per 16x16 output tile; 64 waves total; v_wmma_f32_16x16x4_f32.
// Fragment layout (32-bit, per ISA 7.12.2):
//   A: lane holds A[m][k0], A[m][k0+1], m=lane&15, k0 = kbase + 2*(lane>>4)
//   B: lane holds B[k0][n], B[k0+1][n],  n=lane&15  (staged contiguously)
//   D: vgpr i -> M = i + 8*(lane>>4), N = lane&15
// The compiler software-pipelines the global_load_b64 stream ~20 iterations
// deep against the WMMA chain; no explicit prefetch needed (and SYS-scope
// global_prefetch_b8 would bypass the WGP cache anyway).
// ---------------------------------------------------------------------------
__global__ void k4_wmma_gemm(const float* __restrict__ A, const float* __restrict__ Bst,
                             const float* __restrict__ terma, float* __restrict__ out) {
    int wave  = (blockIdx.x * blockDim.x + threadIdx.x) >> 5;
    int lane  = threadIdx.x & 31;
    int btile = wave >> 2;          // 0..15
    int ptile = wave & 3;           // 0..3
    int m  = lane & 15;
    int hi = lane >> 4;             // 0 or 1

    const float* arow = A + (size_t)(btile * 16 + m) * KD + 2 * hi;
    const float* bptr = Bst + (size_t)(ptile * NKT) * 64 + (size_t)lane * 2;

    v8f acc = {};
    #pragma unroll 4
    for (int kt = 0; kt < NKT; ++kt) {
        v2f a, b;
        a.x = arow[0]; a.y = arow[1];
        b.x = bptr[0]; b.y = bptr[1];
        acc = __builtin_amdgcn_wmma_f32_16x16x4_f32(
            /*neg_a=*/false, a, /*neg_b=*/false, b,
            /*c_mod=*/(short)0, acc, /*reuse_a=*/false, /*reuse_b=*/false);
        arow += 4;
        bptr += 64;
    }

    int n = lane & 15;
    #pragma unroll
    for (int i = 0; i < 8; ++i) {
        int brow = btile * 16 + i + 8 * hi;
        int pcol = ptile * 16 + n;
        out[OFF_LOSS + (unsigned)(brow * CC + pcol)] = acc[i] + terma[brow];
    }
}

// ---------------------------------------------------------------------------
// K5: passthrough outputs + prototype gather + proto_count base copy
// ---------------------------------------------------------------------------
__global__ void k5_emit(const float* __restrict__ x, const int* __restrict__ label,
                        const float* __restrict__ mask, const float* __restrict__ proto,
                        const float* __restrict__ pcount, float* __restrict__ out) {
    int idx = blockIdx.x * blockDim.x + threadIdx.x;
    if (idx < BB * TD) {
        int b = idx / TD, td = idx - b * TD;
        out[OFF_OUTSEQ + idx] = proto[(size_t)label[b] * TD + td];  // output_seq gather
        out[OFF_INSEQ  + idx] = x[idx];                              // input_seq copy
    }
    if (idx < BB) {
        float lf = (float)label[idx];
        out[OFF_IDX + idx] = lf;      // indices == label (K==1)
        out[OFF_LAB + idx] = lf;
    }
    if (idx < BB * TT) out[OFF_MASK + idx] = mask[idx];
    if (idx < CC)      out[OFF_PCOUNT + idx] = pcount[idx];
}

// ---------------------------------------------------------------------------
// K6: proto_count[label[b]] += 1  (after K5's base copy; stream-ordered)
// ---------------------------------------------------------------------------
__global__ void k6_counts(const int* __restrict__ label, float* __restrict__ out) {
    int b = threadIdx.x;
    if (b < BB) atomicAdd(&out[OFF_PCOUNT + label[b]], 1.0f);
}

extern "C" void kernel_launch(void* const* d_in, const int* in_sizes, int n_in,
                              void* d_out, int out_size, void* d_ws, size_t ws_size,
                              hipStream_t stream) {
    const float* x      = (const float*)d_in[0];   // (B,T,D)
    const int*   label  = (const int*)  d_in[1];   // (B,)
    const float* mask   = (const float*)d_in[2];   // (B,T)
    const float* proto  = (const float*)d_in[3];   // (C,T,D)
    const float* w      = (const float*)d_in[4];   // (T,D)
    const float* pcount = (const float*)d_in[5];   // (P,)
    float* out = (float*)d_out;
    float* ws  = (float*)d_ws;

    float* ws_wn    = ws + WS_WN;
    float* ws_terma = ws + WS_TERMA;
    float* ws_A     = ws + WS_A;
    float* ws_B     = ws + WS_B;

    k0_softplus_norm<<<1, 256, 0, stream>>>(w, ws_wn);

    k1_build_a<<<(BB * KD + 255) / 256, 256, 0, stream>>>(x, mask, ws_wn, ws_A);
    k2_build_b<<<(KD * CC + 255) / 256, 256, 0, stream>>>(proto, ws_wn, ws_B);
    k3_term_a<<<BB, 256, 0, stream>>>(x, ws_A, ws_terma);

    // 64 output tiles -> 64 waves -> 8 blocks x 256 threads (8 waves/block)
    k4_wmma_gemm<<<8, 256, 0, stream>>>(ws_A, ws_B, ws_terma, out);

    k5_emit<<<(BB * TD + 255) / 256, 256, 0, stream>>>(x, label, mask, proto, pcount, out);
    k6_counts<<<1, 256, 0, stream>>>(label, out);
}